// RoIPool_59270548685132
// MI455X (gfx1250) — compile-verified
//
#include <hip/hip_runtime.h>
#include <hip/hip_bf16.h>

// RoI max-pool (mmcv semantics) for MI455X / gfx1250.
// input (4,128,56,56) f32, rois (256,5) f32 -> output (256,128,7,7) f32.
//
// Grid = (256 rois, 4 channel-groups), block = 256 threads (8 wave32s):
//  - roi row is block-uniform -> s_load + scalar-FP geometry (s_rndne_f32 etc.)
//  - 49 bin rectangles computed once per block into LDS (ds path)
//  - each block owns a contiguous 32-channel x 49-bin output slice -> writes
//    stay coalesced; 1024 blocks / 8192 waves fill the chip (latency-bound op)
//  - gfx1250 global_prefetch_b8 pulls the next bin row into WGP$ while the
//    current row is max-reduced (feature map is L2-resident: 6.4MB << 192MB)

#define POOLED_H 7
#define POOLED_W 7
#define NBINS    (POOLED_H * POOLED_W)   // 49
#define CHANS    128
#define CSPLIT   4                       // channel groups per roi
#define CGRP     (CHANS / CSPLIT)        // 32 channels per block
#define FEAT_H   56
#define FEAT_W   56
#define PLANE    (FEAT_H * FEAT_W)       // 3136
#define SCALE    0.25f

__global__ __launch_bounds__(256) void roi_pool_fwd_kernel(
    const float* __restrict__ inp,   // (B, C, H, W)
    const float* __restrict__ rois,  // (N, 5) = [b, x1, y1, x2, y2]
    float* __restrict__ out)         // (N, C, 7, 7)
{
    __shared__ int4 bins[NBINS];     // {hstart, hend, wstart, wend}

    const int n   = blockIdx.x;      // roi id      (uniform -> scalarized)
    const int cg  = blockIdx.y;      // channel grp (uniform)
    const int tid = threadIdx.x;
    const float* r = rois + n * 5;

    if (tid < NBINS) {
        const int ph = tid / POOLED_W;
        const int pw = tid % POOLED_W;

        // jnp.round == round-half-to-even == rintf under default RNE mode.
        const int sw = (int)rintf(r[1] * SCALE);
        const int sh = (int)rintf(r[2] * SCALE);
        const int ew = (int)rintf(r[3] * SCALE);
        const int eh = (int)rintf(r[4] * SCALE);

        int roi_w_i = ew - sw + 1; if (roi_w_i < 1) roi_w_i = 1;
        int roi_h_i = eh - sh + 1; if (roi_h_i < 1) roi_h_i = 1;
        const float bin_h = (float)roi_h_i / (float)POOLED_H;
        const float bin_w = (float)roi_w_i / (float)POOLED_W;

        int hstart = (int)floorf(bin_h * (float)ph)       + sh;
        int hend   = (int)ceilf (bin_h * (float)(ph + 1)) + sh;
        int wstart = (int)floorf(bin_w * (float)pw)       + sw;
        int wend   = (int)ceilf (bin_w * (float)(pw + 1)) + sw;
        hstart = min(max(hstart, 0), FEAT_H);
        hend   = min(max(hend,   0), FEAT_H);
        wstart = min(max(wstart, 0), FEAT_W);
        wend   = min(max(wend,   0), FEAT_W);

        bins[tid] = make_int4(hstart, hend, wstart, wend);
    }
    __syncthreads();

    const int b = (int)r[0];                          // uniform scalar load
    const int c0 = cg * CGRP;                         // first channel of this block
    const float* __restrict__ bbase =
        inp + (size_t)(b * CHANS + c0) * PLANE;
    float* __restrict__ obase =
        out + ((size_t)n * CHANS + c0) * NBINS;

    for (int idx = tid; idx < CGRP * NBINS; idx += 256) {
        const int c   = idx / NBINS;                  // cheap mulhi
        const int bin = idx - c * NBINS;
        const int4 bb = bins[bin];                    // ds_load_b128

        float v = 0.0f;                               // mmcv: empty bins yield 0
        if (bb.y > bb.x && bb.w > bb.z) {
            v = -INFINITY;
            const float* __restrict__ base = bbase + (size_t)c * PLANE;
            const int span = bb.w - bb.z;
            for (int h = bb.x; h < bb.y; ++h) {
                const float* __restrict__ row = base + h * FEAT_W + bb.z;
                if (h + 1 < bb.y) {
                    // gfx1250 global_prefetch_b8: next row of this bin into WGP$
                    __builtin_prefetch(base + (h + 1) * FEAT_W + bb.z, 0, 0);
                }
                #pragma unroll 4
                for (int w = 0; w < span; ++w) {
                    v = fmaxf(v, row[w]);
                }
            }
        }
        obase[idx] = v;                               // coalesced slice write
    }
}

extern "C" void kernel_launch(void* const* d_in, const int* in_sizes, int n_in,
                              void* d_out, int out_size, void* d_ws, size_t ws_size,
                              hipStream_t stream) {
    const float* inp  = (const float*)d_in[0];   // (4,128,56,56) f32
    const float* rois = (const float*)d_in[1];   // (256,5) f32
    float* out = (float*)d_out;                  // (256,128,7,7) f32

    const int n_rois = in_sizes[1] / 5;          // 256
    dim3 grid(n_rois, CSPLIT);
    roi_pool_fwd_kernel<<<grid, 256, 0, stream>>>(inp, rois, out);
}